// GAT_66348654789014
// MI455X (gfx1250) — compile-verified
//
#include <hip/hip_runtime.h>
#include <hip/hip_bf16.h>
#include <math.h>

// ---------------------------------------------------------------------------
// Types for CDNA5 WMMA f32 16x16x4:  A,B = 2 VGPRs (v2f), C/D = 8 VGPRs (v8f)
// ---------------------------------------------------------------------------
typedef float v2f __attribute__((ext_vector_type(2)));
typedef float v8f __attribute__((ext_vector_type(8)));

#define HID   512
#define HEADS 4
#define FDIM  2048          // HEADS*HID
#define NEG_SLOPE 0.2f

// ---------------------------------------------------------------------------
// Monotone float<->uint encoding for atomicMax-based segment max
// ---------------------------------------------------------------------------
__device__ __forceinline__ unsigned fenc(float x) {
    unsigned b = __float_as_uint(x);
    return (b & 0x80000000u) ? ~b : (b | 0x80000000u);
}
__device__ __forceinline__ float fdec(unsigned u) {
    unsigned b = (u & 0x80000000u) ? (u & 0x7FFFFFFFu) : ~u;
    return __uint_as_float(b);
}
__device__ __forceinline__ float leaky(float x) {
    return x > 0.0f ? x : NEG_SLOPE * x;
}

// ---------------------------------------------------------------------------
// GEMM: C[M,Nn] = A[M,K] @ B[K,Nn]  (+bias, +relu) via V_WMMA_F32_16X16X4_F32
// One wave computes a 32(M) x 64(N) tile: 2 A fragments x 4 B fragments,
// 8 WMMAs per K=4 step. Out-of-range A rows are CLAMPED (not zeroed): a
// garbage A row only affects the same out-of-range D row, which the store
// masks — keeps the K-loop free of EXEC-mask branches.
// ---------------------------------------------------------------------------
__global__ __launch_bounds__(256)
void gat_gemm_wmma_f32(const float* __restrict__ A, const float* __restrict__ B,
                       float* __restrict__ C, int M, int Nn, int K,
                       const float* __restrict__ bias, int relu)
{
    const int wave   = (int)blockIdx.x * ((int)blockDim.x >> 5) + ((int)threadIdx.x >> 5);
    const int lane   = (int)threadIdx.x & 31;
    const int tilesN = Nn >> 6;                 // 64-wide strips
    const int tilesM = (M + 31) >> 5;           // 32-row tiles
    if (wave >= tilesM * tilesN) return;

    const int mt   = wave / tilesN;
    const int ng   = wave % tilesN;
    const int half = lane >> 4;                 // K pair select / M row-half select
    const int r    = lane & 15;
    const int m0   = mt << 5;
    const int n0   = ng << 6;

    // Clamped A row indices for the two 16-row sub-tiles
    const int rowA0 = min(m0 + r,      M - 1);
    const int rowA1 = min(m0 + 16 + r, M - 1);

    // Induction pointers: A advances 4 floats / step, B advances 4 rows / step
    const float* aptr0 = A + (size_t)rowA0 * K + (half << 1);
    const float* aptr1 = A + (size_t)rowA1 * K + (half << 1);
    const float* bptr  = B + (size_t)(half << 1) * Nn + n0 + r;
    const size_t bstep = (size_t)4 * Nn;

    v8f acc[2][4] = {};                         // [mSub][nTile] 16x16 f32 accumulators

    for (int k = 0; k < K; k += 4) {
        const v2f a0 = *(const v2f*)aptr0;      // A[row][ka], A[row][ka+1] (b64 load)
        const v2f a1 = *(const v2f*)aptr1;
        __builtin_prefetch(bptr + 2 * bstep, 0, 3);   // global_prefetch: B two steps ahead
#pragma unroll
        for (int t = 0; t < 4; ++t) {
            v2f b;
            b.x = bptr[t * 16];                 // B[ka  ][n]
            b.y = bptr[t * 16 + Nn];            // B[ka+1][n]
            acc[0][t] = __builtin_amdgcn_wmma_f32_16x16x4_f32(
                false, a0, false, b, (short)0, acc[0][t], false, false);
            acc[1][t] = __builtin_amdgcn_wmma_f32_16x16x4_f32(
                false, a1, false, b, (short)0, acc[1][t], false, false);
        }
        aptr0 += 4;
        aptr1 += 4;
        bptr  += bstep;
    }

#pragma unroll
    for (int ms = 0; ms < 2; ++ms) {
#pragma unroll
        for (int t = 0; t < 4; ++t) {
#pragma unroll
            for (int v = 0; v < 8; ++v) {
                const int rr = m0 + (ms << 4) + v + (half << 3);
                if (rr < M) {
                    const int cc = n0 + t * 16 + r;
                    float val = acc[ms][t][v];
                    if (bias) val += bias[cc];
                    if (relu) val = fmaxf(val, 0.0f);
                    C[(size_t)rr * Nn + cc] = val;
                }
            }
        }
    }
}

// ---------------------------------------------------------------------------
// alpha_s[n,h] = dot(xp[n,h,:], a_src[h,:]);  alpha_d likewise.
// One wave per (n,h) pair; wave32 shuffle reduction.
// ---------------------------------------------------------------------------
__global__ __launch_bounds__(256)
void gat_alpha(const float* __restrict__ xp,
               const float* __restrict__ a_src, const float* __restrict__ a_dst,
               float* __restrict__ as, float* __restrict__ ad, int Nn)
{
    const int gw   = ((int)blockIdx.x * (int)blockDim.x + (int)threadIdx.x) >> 5;
    const int lane = (int)threadIdx.x & 31;
    if (gw >= Nn * HEADS) return;
    const int n = gw >> 2;
    const int h = gw & 3;

    const float* xr  = xp    + (size_t)n * FDIM + h * HID;
    const float* asr = a_src + h * HID;
    const float* adr = a_dst + h * HID;

    float s1 = 0.0f, s2 = 0.0f;
    for (int c = lane; c < HID; c += 32) {
        const float xv = xr[c];
        s1 += xv * asr[c];
        s2 += xv * adr[c];
    }
#pragma unroll
    for (int off = 16; off >= 1; off >>= 1) {
        s1 += __shfl_down(s1, off, 32);
        s2 += __shfl_down(s2, off, 32);
    }
    if (lane == 0) {
        as[n * HEADS + h] = s1;
        ad[n * HEADS + h] = s2;
    }
}

// ---------------------------------------------------------------------------
// Init kernels
// ---------------------------------------------------------------------------
__global__ void gat_init_maxsum(unsigned* __restrict__ menc, float* __restrict__ ssum, int total)
{
    const int i = (int)blockIdx.x * (int)blockDim.x + (int)threadIdx.x;
    if (i < total) { menc[i] = 0u; ssum[i] = 0.0f; }
}

__global__ void gat_init_bias_rows(float* __restrict__ out, const float* __restrict__ bias,
                                   int Nn, int ld)
{
    const long long i = (long long)blockIdx.x * blockDim.x + threadIdx.x;
    if (i < (long long)Nn * ld) out[i] = bias[(int)(i % ld)];
}

// ---------------------------------------------------------------------------
// Edge pass 1: segment max of leaky-relu logits over destination segments
// ---------------------------------------------------------------------------
__global__ void gat_edge_max(const int* __restrict__ ei, int E, int Nn,
                             const float* __restrict__ as, const float* __restrict__ ad,
                             unsigned* __restrict__ menc)
{
    const int e = (int)blockIdx.x * (int)blockDim.x + (int)threadIdx.x;
    const int Et = E + Nn;
    if (e >= Et) return;
    int s, d;
    if (e < E) { s = ei[e]; d = ei[E + e]; } else { s = d = e - E; }
#pragma unroll
    for (int h = 0; h < HEADS; ++h) {
        const float l = leaky(as[s * HEADS + h] + ad[d * HEADS + h]);
        atomicMax(&menc[d * HEADS + h], fenc(l));
    }
}

// ---------------------------------------------------------------------------
// Edge pass 2: e = exp(logit - m[dst]); store per-edge, segment-sum into ssum
// ---------------------------------------------------------------------------
__global__ void gat_edge_expsum(const int* __restrict__ ei, int E, int Nn,
                                const float* __restrict__ as, const float* __restrict__ ad,
                                const unsigned* __restrict__ menc,
                                float* __restrict__ ew, float* __restrict__ ssum)
{
    const int e = (int)blockIdx.x * (int)blockDim.x + (int)threadIdx.x;
    const int Et = E + Nn;
    if (e >= Et) return;
    int s, d;
    if (e < E) { s = ei[e]; d = ei[E + e]; } else { s = d = e - E; }
#pragma unroll
    for (int h = 0; h < HEADS; ++h) {
        const float l = leaky(as[s * HEADS + h] + ad[d * HEADS + h]);
        const float w = __expf(l - fdec(menc[d * HEADS + h]));
        ew[e * HEADS + h] = w;
        atomicAdd(&ssum[d * HEADS + h], w);
    }
}

// ---------------------------------------------------------------------------
// Aggregation: out[dst,:] += xp[src,:] * (ew[e,h] / ssum[dst,h])
// One 256-thread block per edge; 8 channels/thread; f32 global atomic adds.
// ---------------------------------------------------------------------------
__global__ __launch_bounds__(256)
void gat_aggregate(const int* __restrict__ ei, int E, int Nn,
                   const float* __restrict__ xp,
                   const float* __restrict__ ew, const float* __restrict__ ssum,
                   float* __restrict__ out)
{
    const int e = (int)blockIdx.x;
    int s, d;
    if (e < E) { s = ei[e]; d = ei[E + e]; } else { s = d = e - E; }

    __shared__ float w[HEADS];
    if (threadIdx.x < HEADS)
        w[threadIdx.x] = ew[e * HEADS + threadIdx.x] / ssum[d * HEADS + threadIdx.x];
    __syncthreads();

    const float* xr   = xp  + (size_t)s * FDIM;
    float*       orow = out + (size_t)d * FDIM;
#pragma unroll
    for (int j = 0; j < FDIM / 256; ++j) {
        const int i = (int)threadIdx.x + 256 * j;
        atomicAdd(&orow[i], xr[i] * w[i >> 9]);   // i>>9 == channel/512 == head
    }
}

// ---------------------------------------------------------------------------
// Host side
// ---------------------------------------------------------------------------
static inline int cdiv(int a, int b) { return (a + b - 1) / b; }

static void run_gemm(const float* A, const float* B, float* C, int M, int Nn, int K,
                     const float* bias, int relu, hipStream_t stream)
{
    const int tiles = cdiv(M, 32) * (Nn / 64);
    gat_gemm_wmma_f32<<<cdiv(tiles, 8), 256, 0, stream>>>(A, B, C, M, Nn, K, bias, relu);
}

extern "C" void kernel_launch(void* const* d_in, const int* in_sizes, int n_in,
                              void* d_out, int out_size, void* d_ws, size_t ws_size,
                              hipStream_t stream)
{
    const float* x      = (const float*)d_in[0];
    const int*   ei     = (const int*)  d_in[1];   // [2,E] flat: row0=src, row1=dst
    const float* W1     = (const float*)d_in[3];
    const float* a_src1 = (const float*)d_in[4];
    const float* a_dst1 = (const float*)d_in[5];
    const float* b1     = (const float*)d_in[6];
    const float* W2     = (const float*)d_in[7];
    const float* a_src2 = (const float*)d_in[8];
    const float* a_dst2 = (const float*)d_in[9];
    const float* b2     = (const float*)d_in[10];
    const float* Wl     = (const float*)d_in[11];
    const float* bl     = (const float*)d_in[12];
    float* out = (float*)d_out;

    const int N  = in_sizes[0] / 128;   // 10000
    const int IN = 128;
    const int E  = in_sizes[1] / 2;     // 160000
    const int Et = E + N;               // edges + self loops
    const int OUT = 128;

    // Workspace layout (floats)
    float* ws   = (float*)d_ws;
    float* xp   = ws;                               // [N, 2048]  xp1 -> xp2
    float* hb   = xp + (size_t)N * FDIM;            // [N, 2048]  h1  -> h2
    float* as_  = hb + (size_t)N * FDIM;            // [N, 4]
    float* ad_  = as_ + (size_t)N * HEADS;          // [N, 4]
    unsigned* menc = (unsigned*)(ad_ + (size_t)N * HEADS);   // [N, 4]
    float* ssum = (float*)(menc + (size_t)N * HEADS);        // [N, 4]
    float* ew   = ssum + (size_t)N * HEADS;         // [Et, 4]

    const int TB = 256;
    const int nhWaves = cdiv(N * HEADS, 8);         // 8 waves per 256-thread block

    // ---------------- Layer 1 ----------------
    run_gemm(x, W1, xp, N, FDIM, IN, nullptr, 0, stream);                 // xp1 = x @ W1
    gat_alpha<<<nhWaves, TB, 0, stream>>>(xp, a_src1, a_dst1, as_, ad_, N);
    gat_init_maxsum<<<cdiv(N * HEADS, TB), TB, 0, stream>>>(menc, ssum, N * HEADS);
    gat_init_bias_rows<<<cdiv(N * FDIM, TB), TB, 0, stream>>>(hb, b1, N, FDIM);
    gat_edge_max   <<<cdiv(Et, TB), TB, 0, stream>>>(ei, E, N, as_, ad_, menc);
    gat_edge_expsum<<<cdiv(Et, TB), TB, 0, stream>>>(ei, E, N, as_, ad_, menc, ew, ssum);
    gat_aggregate  <<<Et, TB, 0, stream>>>(ei, E, N, xp, ew, ssum, hb);   // hb = h1

    // ---------------- Layer 2 ----------------
    run_gemm(hb, W2, xp, N, FDIM, FDIM, nullptr, 0, stream);              // xp2 = h1 @ W2
    gat_alpha<<<nhWaves, TB, 0, stream>>>(xp, a_src2, a_dst2, as_, ad_, N);
    gat_init_maxsum<<<cdiv(N * HEADS, TB), TB, 0, stream>>>(menc, ssum, N * HEADS);
    gat_init_bias_rows<<<cdiv(N * FDIM, TB), TB, 0, stream>>>(hb, b2, N, FDIM);
    gat_edge_max   <<<cdiv(Et, TB), TB, 0, stream>>>(ei, E, N, as_, ad_, menc);
    gat_edge_expsum<<<cdiv(Et, TB), TB, 0, stream>>>(ei, E, N, as_, ad_, menc, ew, ssum);
    gat_aggregate  <<<Et, TB, 0, stream>>>(ei, E, N, xp, ew, ssum, hb);   // hb = h2

    // ---------------- Linear head + ReLU ----------------
    run_gemm(hb, Wl, out, N, OUT, FDIM, bl, 1, stream);                   // relu(h2@Wl + bl)
}